// TemplateLayer_80753975099715
// MI455X (gfx1250) — compile-verified
//
#include <hip/hip_runtime.h>
#include <hip/hip_bf16.h>
#include <math.h>

typedef __attribute__((ext_vector_type(2))) float v2f;
typedef __attribute__((ext_vector_type(8))) float v8f;

#define NE_CONST 150000

// C[M x 128] = A[M x 128] @ W[128 x 128], all f32 row-major.
// One block = 256 threads = 8 waves; block covers 128 rows, each wave a
// 16-row strip across all 128 output columns (8 C-tiles of 16x16).
// V_WMMA_F32_16X16X4_F32, K-step 4. W is staged in LDS K-interleaved so each
// B fragment is one conflict-free ds_load_b64.
__global__ __launch_bounds__(256) void gemm128_wmma(
    const float* __restrict__ A, const float* __restrict__ W,
    float* __restrict__ C, int M)
{
    // K-interleaved W: element (k, n) at float index (k>>2)*512 + n*4 + (k&3)
    __shared__ float sW[128 * 128];   // 64 KB

    const int tid  = threadIdx.x;
    const int wave = tid >> 5;        // 0..7
    const int lane = tid & 31;
    const int blockRow = blockIdx.x * 128;

    // Cooperative swizzling load of W (coalesced global float4 reads)
    #pragma unroll
    for (int i = tid; i < 128 * 128 / 4; i += 256) {
        const float4 w = ((const float4*)W)[i];
        const int k  = i >> 5;          // row (K)
        const int n0 = (i & 31) << 2;   // col (N)
        float* p = sW + (k >> 2) * 512 + (k & 3);
        p[(n0 + 0) * 4] = w.x;
        p[(n0 + 1) * 4] = w.y;
        p[(n0 + 2) * 4] = w.z;
        p[(n0 + 3) * 4] = w.w;
    }
    __syncthreads();

    // A fragment addressing (ISA 32-bit A 16x4 layout):
    //   M = lane%16; lanes 0-15 hold K = k0+0/k0+1, lanes 16-31 K = k0+2/k0+3
    const int mrow = blockRow + wave * 16 + (lane & 15);
    const int mclamp = (mrow < M) ? mrow : (M - 1);   // clamp: branch-free loads
    const float* arow = A + (size_t)mclamp * 128;
    const int koff = (lane < 16) ? 0 : 2;
    const int ncol = lane & 15;

    v8f acc[8];
    #pragma unroll
    for (int t = 0; t < 8; ++t) acc[t] = (v8f){0,0,0,0,0,0,0,0};

    const float* bbase = sW + ncol * 4 + koff;

    // Software-pipelined K loop: A load one step ahead of the 8 WMMAs.
    v2f a = *(const v2f*)(arow + koff);
    for (int k0 = 0; k0 < 128; k0 += 4) {
        const v2f a_cur = a;
        if (k0 + 4 < 128) a = *(const v2f*)(arow + k0 + 4 + koff);
        const float* bp = bbase + (k0 >> 2) * 512;
        #pragma unroll
        for (int t = 0; t < 8; ++t) {
            const v2f b = *(const v2f*)(bp + t * 64);   // ds_load_b64
            acc[t] = __builtin_amdgcn_wmma_f32_16x16x4_f32(
                false, a_cur, false, b, (short)0, acc[t], false, false);
        }
    }

    // C/D 16x16 f32 layout: VGPR v -> M = v (+8 for upper half-wave), N = lane%16
    const int crow0 = blockRow + wave * 16 + ((lane < 16) ? 0 : 8);
    #pragma unroll
    for (int t = 0; t < 8; ++t) {
        #pragma unroll
        for (int v = 0; v < 8; ++v) {
            const int gr = crow0 + v;
            if (gr < M) C[(size_t)gr * 128 + t * 16 + ncol] = acc[t][v];
        }
    }
}

// One wave per nonzero: agg[dst] += xm[src] over 128 channels, deg[dst] += 1.
__global__ __launch_bounds__(256) void scatter_add_kernel(
    const float* __restrict__ xm, const int* __restrict__ src,
    const int* __restrict__ dst, float* __restrict__ agg,
    float* __restrict__ deg, int nnz)
{
    const int w = (int)((blockIdx.x * (unsigned)blockDim.x + threadIdx.x) >> 5);
    const int lane = threadIdx.x & 31;
    if (w >= nnz) return;
    const int s = src[w];
    const int d = dst[w];
    const float* xr = xm + (size_t)s * 128;
    float* ar = agg + (size_t)d * 128;
    #pragma unroll
    for (int c = 0; c < 4; ++c)
        atomicAdd(ar + lane + 32 * c, xr[lane + 32 * c]);
    if (lane == 0) atomicAdd(deg + d, 1.0f);
}

// buf[r, c] = sigmoid(buf[r, c] / deg[r])   (in place)
__global__ __launch_bounds__(256) void sigmoid_div_kernel(
    float* __restrict__ buf, const float* __restrict__ deg, int n_rows)
{
    const int i = blockIdx.x * blockDim.x + threadIdx.x;
    const int total = n_rows * 128;
    if (i < total) {
        const int r = i >> 7;
        const float v = buf[i] / deg[r];
        buf[i] = 1.0f / (1.0f + expf(-v));
    }
}

extern "C" void kernel_launch(void* const* d_in, const int* in_sizes, int n_in,
                              void* d_out, int out_size, void* d_ws, size_t ws_size,
                              hipStream_t stream) {
    const float* x  = (const float*)d_in[0];   // [NF, 128]
    const float* W1 = (const float*)d_in[1];   // [128, 128]
    const float* W2 = (const float*)d_in[2];   // [128, 128]
    const int* rows = (const int*)d_in[3];     // [NNZ] edge ids
    const int* cols = (const int*)d_in[4];     // [NNZ] face ids
    float* out = (float*)d_out;                // [NF, 128]

    const int NF  = in_sizes[0] / 128;
    const int NNZ = in_sizes[3];
    const int NE  = NE_CONST;

    // Workspace layout (f32)
    float* ws    = (float*)d_ws;
    float* xm1   = ws;                          // NF*128
    float* agg_e = xm1 + (size_t)NF * 128;      // NE*128 (becomes x_edges)
    float* xm2   = agg_e + (size_t)NE * 128;    // NE*128
    float* deg_e = xm2 + (size_t)NE * 128;      // NE
    float* deg_f = deg_e + NE;                  // NF

    // Zero accumulators (graph-capturable async memsets)
    hipMemsetAsync(agg_e, 0, (size_t)NE * 128 * sizeof(float), stream);
    hipMemsetAsync(deg_e, 0, (size_t)NE * sizeof(float), stream);
    hipMemsetAsync(deg_f, 0, (size_t)NF * sizeof(float), stream);
    hipMemsetAsync(out,   0, (size_t)NF * 128 * sizeof(float), stream);

    // Level 1: faces -> edges
    gemm128_wmma<<<(NF + 127) / 128, 256, 0, stream>>>(x, W1, xm1, NF);
    scatter_add_kernel<<<(NNZ * 32 + 255) / 256, 256, 0, stream>>>(
        xm1, cols, rows, agg_e, deg_e, NNZ);
    sigmoid_div_kernel<<<((size_t)NE * 128 + 255) / 256, 256, 0, stream>>>(
        agg_e, deg_e, NE);

    // Level 2: edges -> faces
    gemm128_wmma<<<(NE + 127) / 128, 256, 0, stream>>>(agg_e, W2, xm2, NE);
    scatter_add_kernel<<<(NNZ * 32 + 255) / 256, 256, 0, stream>>>(
        xm2, rows, cols, out, deg_f, NNZ);
    sigmoid_div_kernel<<<((size_t)NF * 128 + 255) / 256, 256, 0, stream>>>(
        out, deg_f, NF);
}